// ComputeNodeAreaFromRouteMap_46755013984470
// MI455X (gfx1250) — compile-verified
//
#include <hip/hip_runtime.h>

// Problem constants from the reference (compile-time folded):
// XL = YL = 0.0, XH = YH = 512.0, NBX = NBY = 512, BSX = BSY = 1.0
#define NBX 512
#define NBY 512

// out = 0; for each j: i = idx[j]; out[i] = bilinear-overlap area of cell i
// against utilization_map. Bandwidth-bound streaming (~120 MB/call -> ~5.2 us
// at 23.3 TB/s) + 4-point gather into the L2-resident 1 MB map. No matrix
// structure -> no WMMA by design; wave32 coalesced VMEM is the optimal path.
__global__ __launch_bounds__(256) void ComputeNodeAreaFromRouteMap_kernel(
    const float* __restrict__ pos,        // [2N]  x then y
    const float* __restrict__ node_sx,    // [N]
    const float* __restrict__ node_sy,    // [N]
    const float* __restrict__ umap,       // [NBX*NBY] row-major (bx, by)
    const int*   __restrict__ idx,        // [n_idx]
    float* __restrict__ out,              // [N], pre-zeroed
    int n_idx, int N)
{
    const int j = blockIdx.x * blockDim.x + threadIdx.x;
    if (j >= n_idx) return;               // one element per lane; no div preamble

    const int i = idx[j];

    const float x  = pos[i];
    const float y  = pos[N + i];
    const float xh = x + node_sx[i];
    const float yh = y + node_sy[i];

    // BSX = BSY = 1, XL = YL = 0  ->  bin = clamp(floor(coord), 0, 511)
    const int bx0 = min(max((int)floorf(x), 0), NBX - 1);
    const int by0 = min(max((int)floorf(y), 0), NBY - 1);
    const int bx1 = bx0 + 1;              // may be 512 (weight forced to 0)
    const int by1 = by0 + 1;

    // Overlap weights. Low legs (bx0/by0) are always in range after the clamp;
    // only the +1 legs need the out-of-grid zeroing and index clamp.
    const float f_bx0 = (float)bx0, f_by0 = (float)by0;
    const float f_bx1 = (float)bx1, f_by1 = (float)by1;

    const float ox0 = fmaxf(fminf(xh, f_bx0 + 1.0f) - fmaxf(x, f_bx0), 0.0f);
    const float oy0 = fmaxf(fminf(yh, f_by0 + 1.0f) - fmaxf(y, f_by0), 0.0f);
    float ox1 = fmaxf(fminf(xh, f_bx1 + 1.0f) - fmaxf(x, f_bx1), 0.0f);
    float oy1 = fmaxf(fminf(yh, f_by1 + 1.0f) - fmaxf(y, f_by1), 0.0f);
    ox1 = (bx1 < NBX) ? ox1 : 0.0f;
    oy1 = (by1 < NBY) ? oy1 : 0.0f;

    const int bxc1 = min(bx1, NBX - 1);
    const int byc1 = min(by1, NBY - 1);

    const float* __restrict__ r0 = umap + bx0  * NBY;
    const float* __restrict__ r1 = umap + bxc1 * NBY;

    // Four L2-hit gathers; accumulate in the reference order
    // (dx0,dy0) + (dx0,dy1) + (dx1,dy0) + (dx1,dy1).
    float area =            ox0 * oy0 * r0[by0];
    area = fmaf(ox0 * oy1, r0[byc1], area);
    area = fmaf(ox1 * oy0, r1[by0],  area);
    area = fmaf(ox1 * oy1, r1[byc1], area);

    out[i] = area;
}

extern "C" void kernel_launch(void* const* d_in, const int* in_sizes, int n_in,
                              void* d_out, int out_size, void* d_ws, size_t ws_size,
                              hipStream_t stream) {
    // setup_inputs() order:
    //   0: pos (2N f32), 1: node_size_x (N f32), 2: node_size_y (N f32),
    //   3: utilization_map (512*512 f32), 4: flop_lut_indices (N i32)
    const float* pos  = (const float*)d_in[0];
    const float* nsx  = (const float*)d_in[1];
    const float* nsy  = (const float*)d_in[2];
    const float* umap = (const float*)d_in[3];
    const int*   idx  = (const int*)d_in[4];
    float* out = (float*)d_out;

    const int N     = in_sizes[1];   // node count
    const int n_idx = in_sizes[4];   // number of scatter indices

    // General scatter semantics: out starts at zero every call
    // (graph-capture-safe memset node).
    hipMemsetAsync(d_out, 0, (size_t)out_size * sizeof(float), stream);

    const int threads = 256;                       // 8 wave32s per block
    const int blocks  = (n_idx + threads - 1) / threads;
    ComputeNodeAreaFromRouteMap_kernel<<<blocks, threads, 0, stream>>>(
        pos, nsx, nsy, umap, idx, out, n_idx, N);
}